// LSTMStacked_18648747999256
// MI455X (gfx1250) — compile-verified
//
#include <hip/hip_runtime.h>
#include <math.h>

typedef __attribute__((ext_vector_type(2))) float v2f;
typedef __attribute__((ext_vector_type(8))) float v8f;

#define NB 16384   // batch
#define NT 512     // timesteps
#define NL 4       // layers
#define NH 2       // hidden
#define NTOT (NB * NT * NH)  // 16,777,216 output elements
#define PFD 4      // label prefetch depth (register pipeline)

__device__ __forceinline__ float tanh_fast(float x) {
#if __has_builtin(__builtin_amdgcn_tanhf)
    return __builtin_amdgcn_tanhf(x);
#elif __has_builtin(__builtin_amdgcn_tanh_f32)
    return __builtin_amdgcn_tanh_f32(x);
#else
    return tanhf(x);
#endif
}

// One wave = 16 batch elements. Per layer-step, one v_wmma_f32_16x16x4_f32:
//   A (16x4): rows 0..7 = [W_ih | W_hh] gate rows, rows 8..15 = DUPLICATE of
//             rows 0..7 (so D's upper half hands high lanes the same gates).
//             Sigmoid-gate rows (i,f,o) are pre-scaled by 0.5 so that
//             sigma(x) = 0.5*tanh(x/2)+0.5 needs no pre-multiply.
//   B (4x16): 16 batch columns; K=(x0,x1) from low lanes (layer input),
//             K=(h0,h1) from high lanes (layer recurrent state).
//   C       : bias broadcast (same pre-scaling as A rows), pinned in VGPRs.
// D gives BOTH lane m and lane m+16 all 8 gates of element m, so every lane
// updates c/h locally -> zero cross-lane ops in the recurrence.
// Labels are consumed through a depth-4 register pipeline so the global-load
// wait never lands on the recurrence critical path.
__global__ void __launch_bounds__(32) lstm_wmma_kernel(
    const float* __restrict__ xin,   // [B,2]
    const float* __restrict__ lab,   // [B,T,2]
    const float* __restrict__ Wih,   // [L,8,2]
    const float* __restrict__ Whh,   // [L,8,2]
    const float* __restrict__ bhh,   // [L,8]
    float* __restrict__ ret,         // [B,T,2]
    float* __restrict__ partials)    // [gridDim.x]
{
    const int lane = threadIdx.x & 31;
    const bool lo  = lane < 16;
    const int  m   = lane & 15;                 // batch sub-index (column)
    const int  r   = lane & 7;                  // gate row feeding this lane's A
    const int  bglob = blockIdx.x * 16 + m;

    // ---- A matrices (weights) and C (bias), per layer ----
    v2f A[NL];
    v8f Cb[NL];
#pragma unroll
    for (int l = 0; l < NL; ++l) {
        // low lanes supply K=0,1 (W_ih), high lanes K=2,3 (W_hh)
        const float* W = lo ? (Wih + l * 16 + r * 2)
                            : (Whh + l * 16 + r * 2);
        // gate order i(0,1) f(2,3) g(4,5) o(6,7): scale sigmoid rows by 0.5
        const float s = (r == 4 || r == 5) ? 1.0f : 0.5f;
        A[l].x = s * W[0];
        A[l].y = s * W[1];
#pragma unroll
        for (int q = 0; q < 8; ++q) {
            const float sq = (q == 4 || q == 5) ? 1.0f : 0.5f;
            float tmp = sq * bhh[l * 8 + q];
            // Pin into a VGPR: the bias is wave-uniform, and without this the
            // compiler keeps it in SGPRs and re-materializes 8 VGPRs with
            // v_mov_b64 before every WMMA (SRC2 must be VGPRs).
            asm("" : "+v"(tmp));
            Cb[l][q] = tmp;
        }
    }

    // ---- recurrent state (identical trajectories in both wave halves) ----
    float hs0[NL], hs1[NL], c0[NL], c1[NL];
#pragma unroll
    for (int l = 0; l < NL; ++l) { hs0[l] = hs1[l] = c0[l] = c1[l] = 0.0f; }

    float x0 = xin[bglob * 2 + 0];
    float x1 = xin[bglob * 2 + 1];
    float acc = 0.0f;

    const float2* labp = (const float2*)lab + (size_t)bglob * NT;
    float2*       retp = (float2*)ret       + (size_t)bglob * NT;

    // ---- label prefetch pipeline (depth PFD) ----
    float2 lb[PFD];
#pragma unroll
    for (int k = 0; k < PFD; ++k)
        lb[k] = labp[k];

#pragma unroll 4
    for (int t = 0; t < NT; ++t) {
        // consume this step's label; immediately refill the slot with t+PFD
        // (address wraps via mask at the tail of the sequence; value unused)
        const float2 lb_cur = lb[t & (PFD - 1)];
        lb[t & (PFD - 1)] = labp[(t + PFD) & (NT - 1)];

#pragma unroll
        for (int l = 0; l < NL; ++l) {
            // B matrix: low lanes carry layer input, high lanes carry h state
            v2f Bm;
            Bm.x = lo ? x0 : hs0[l];
            Bm.y = lo ? x1 : hs1[l];

            v8f g = __builtin_amdgcn_wmma_f32_16x16x4_f32(
                false, A[l], false, Bm, (short)0, Cb[l], false, false);

            // g[0..3], g[6..7] already hold 0.5*preact (sigmoid rows);
            // g[4..5] hold full preact (tanh rows)
            float i0 = __builtin_fmaf(0.5f, tanh_fast(g[0]), 0.5f);
            float i1 = __builtin_fmaf(0.5f, tanh_fast(g[1]), 0.5f);
            float f0 = __builtin_fmaf(0.5f, tanh_fast(g[2]), 0.5f);
            float f1 = __builtin_fmaf(0.5f, tanh_fast(g[3]), 0.5f);
            float t0 = tanh_fast(g[4]);
            float t1 = tanh_fast(g[5]);
            float o0 = __builtin_fmaf(0.5f, tanh_fast(g[6]), 0.5f);
            float o1 = __builtin_fmaf(0.5f, tanh_fast(g[7]), 0.5f);

            c0[l] = __builtin_fmaf(f0, c0[l], i0 * t0);
            c1[l] = __builtin_fmaf(f1, c1[l], i1 * t1);
            float h0 = o0 * tanh_fast(c0[l]);
            float h1 = o1 * tanh_fast(c1[l]);

            // every lane computed h locally: high lanes keep it as layer
            // state, low lanes pass it on as next layer's input
            hs0[l] = h0;
            hs1[l] = h1;
            x0 = h0;
            x1 = h1;
        }
        if (lo) {
            float2 o; o.x = x0; o.y = x1;
            retp[t] = o;
            float d0 = x0 - lb_cur.x;
            float d1 = x1 - lb_cur.y;
            acc = __builtin_fmaf(d0, d0, acc);
            acc = __builtin_fmaf(d1, d1, acc);
        }
    }

    // deterministic wave reduction (high lanes contribute 0)
#pragma unroll
    for (int off = 16; off >= 1; off >>= 1)
        acc += __shfl_xor(acc, off, 32);
    if (lane == 0)
        partials[blockIdx.x] = acc;
}

__global__ void __launch_bounds__(256) loss_reduce_kernel(
    const float* __restrict__ partials, int n, float* __restrict__ out_loss)
{
    __shared__ double sm[256];
    double s = 0.0;
    for (int i = threadIdx.x; i < n; i += 256)
        s += (double)partials[i];
    sm[threadIdx.x] = s;
    __syncthreads();
    for (int stride = 128; stride >= 1; stride >>= 1) {
        if ((int)threadIdx.x < stride)
            sm[threadIdx.x] += sm[threadIdx.x + stride];
        __syncthreads();
    }
    if (threadIdx.x == 0)
        *out_loss = (float)(sm[0] / (double)NTOT);
}

extern "C" void kernel_launch(void* const* d_in, const int* in_sizes, int n_in,
                              void* d_out, int out_size, void* d_ws, size_t ws_size,
                              hipStream_t stream) {
    const float* xin = (const float*)d_in[0];  // inputs [1,B,2]
    const float* lab = (const float*)d_in[1];  // labels [1,B,T,2]
    const float* Wih = (const float*)d_in[2];  // [L,8,2]
    const float* Whh = (const float*)d_in[3];  // [L,8,2]
    const float* bhh = (const float*)d_in[4];  // [L,8]

    float* out      = (float*)d_out;           // ret flat, then loss scalar
    float* partials = (float*)d_ws;

    const int nblocks = NB / 16;               // 1024 single-wave blocks

    lstm_wmma_kernel<<<nblocks, 32, 0, stream>>>(
        xin, lab, Wih, Whh, bhh, out, partials);

    loss_reduce_kernel<<<1, 256, 0, stream>>>(
        partials, nblocks, out + (size_t)NTOT);
}